// CWG_82214263980322
// MI455X (gfx1250) — compile-verified
//
#include <hip/hip_runtime.h>
#include <hip/hip_bf16.h>

// Problem constants (match reference)
#define BATCH   32
#define CH      2048
#define SIZE    4096

#define TPB     256          // threads per block (8 waves of 32)
#define BB      8            // batches per block (register accumulators)
#define CH_STEP 8            // channels staged per LDS buffer
#define S_PER_BLOCK (TPB*4)  // 1024 s-values per block (float4 per thread)
#define CSPLIT_MAX 16

typedef float v4f __attribute__((ext_vector_type(4)));
typedef int   v4i __attribute__((ext_vector_type(4)));

// Pointer types matching the builtin's signature (probe-confirmed round 1):
//   param0: int4 in global memory (AS1), param1: int4 in LDS (AS3)
typedef __attribute__((address_space(1))) v4i* gv4i_p;
typedef __attribute__((address_space(3))) v4i* lv4i_p;

// ---------------------------------------------------------------------------
// CDNA5 async global->LDS copy (ASYNCcnt-tracked) + wait.
// ---------------------------------------------------------------------------
__device__ __forceinline__ void async_load_b128(const float* g, float* l) {
#if __has_builtin(__builtin_amdgcn_global_load_async_to_lds_b128)
  __builtin_amdgcn_global_load_async_to_lds_b128((gv4i_p)(g), (lv4i_p)(l), 0, 0);
#else
  unsigned lds_off = (unsigned)(unsigned long long)
      ((__attribute__((address_space(3))) void*)(l));
  asm volatile("global_load_async_to_lds_b128 %0, %1, off"
               :: "v"(lds_off), "v"(g) : "memory");
#endif
}

#if __has_builtin(__builtin_amdgcn_s_wait_asynccnt)
#define WAIT_ASYNC(n) __builtin_amdgcn_s_wait_asynccnt(n)
#else
#define WAIT_ASYNC(n) asm volatile("s_wait_asynccnt " #n ::: "memory")
#endif

// ---------------------------------------------------------------------------
// Main kernel: each block handles (s-tile of 1024, 8 batches, cpb channels).
// W chunk is double-buffered through LDS via async copies; x is streamed with
// non-temporal b128 loads (zero reuse -> keep L2 for W).
// ---------------------------------------------------------------------------
__global__ __launch_bounds__(TPB) void diag_dot_kernel(
    const float* __restrict__ x, const float* __restrict__ w,
    const float* __restrict__ bias, float* __restrict__ dst,
    int cpb, int finalize)
{
  __shared__ float sm[2][CH_STEP][S_PER_BLOCK];   // 64 KiB

  const int t  = threadIdx.x;
  const int s  = blockIdx.x * S_PER_BLOCK + t * 4;
  const int b0 = blockIdx.y * BB;
  const int c0 = blockIdx.z * cpb;
  const int nchunk = cpb / CH_STEP;

  const float* wbase = w + (size_t)c0 * SIZE + s;
  const float* xbase = x + ((size_t)b0 * CH + (size_t)c0) * SIZE + s;

  v4f acc[BB];
#pragma unroll
  for (int i = 0; i < BB; ++i) acc[i] = 0.0f;

  // Prologue: stage chunk 0 into buffer 0 (8 async b128 ops per wave)
#pragma unroll
  for (int ch = 0; ch < CH_STEP; ++ch)
    async_load_b128(wbase + (size_t)ch * SIZE, &sm[0][ch][t * 4]);

  for (int k = 0; k < nchunk; ++k) {
    const int buf = k & 1;
    if (k + 1 < nchunk) {
      const float* wp = wbase + (size_t)(k + 1) * CH_STEP * SIZE;
#pragma unroll
      for (int ch = 0; ch < CH_STEP; ++ch)
        async_load_b128(wp + (size_t)ch * SIZE, &sm[buf ^ 1][ch][t * 4]);
      WAIT_ASYNC(8);   // in-order completion: current chunk's 8 are done
    } else {
      WAIT_ASYNC(0);
    }
    __syncthreads();   // chunk k resident for every wave in the block

    const float* xck = xbase + (size_t)k * CH_STEP * SIZE;
#pragma unroll 4
    for (int ch = 0; ch < CH_STEP; ++ch) {
      v4f w4 = *(const v4f*)&sm[buf][ch][t * 4];       // ds_load_b128
      const float* xc = xck + (size_t)ch * SIZE;
#pragma unroll
      for (int bi = 0; bi < BB; ++bi) {
        v4f xv = __builtin_nontemporal_load(
            (const v4f*)(xc + (size_t)bi * CH * SIZE)); // NT b128 stream
        acc[bi] += xv * w4;                             // fused to v_fma
      }
    }
    __syncthreads();   // all waves done reading buf before it is reloaded
  }

  if (finalize) {
    v4f bz = *(const v4f*)(bias + s);
#pragma unroll
    for (int bi = 0; bi < BB; ++bi) {
      v4f r = acc[bi] + bz;
#pragma unroll
      for (int j = 0; j < 4; ++j) r[j] = fmaxf(r[j], 0.0f);
      *(v4f*)(dst + (size_t)(b0 + bi) * SIZE + s) = r;
    }
  } else {
    float* p = dst + ((size_t)blockIdx.z * BATCH + b0) * SIZE + s;
#pragma unroll
    for (int bi = 0; bi < BB; ++bi)
      *(v4f*)(p + (size_t)bi * SIZE) = acc[bi];
  }
}

// ---------------------------------------------------------------------------
// Reduction over channel-splits + bias + ReLU (tiny: 8 MiB traffic)
// ---------------------------------------------------------------------------
__global__ __launch_bounds__(256) void reduce_kernel(
    const float* __restrict__ part, const float* __restrict__ bias,
    float* __restrict__ out, int csplit)
{
  const int i = blockIdx.x * 256 + threadIdx.x;   // over BATCH*SIZE/4
  const int s = (i * 4) & (SIZE - 1);
  const int b = (i * 4) / SIZE;
  v4f sum = 0.0f;
  for (int k = 0; k < csplit; ++k)
    sum += *(const v4f*)(part + ((size_t)k * BATCH + b) * SIZE + s);
  sum += *(const v4f*)(bias + s);
#pragma unroll
  for (int j = 0; j < 4; ++j) sum[j] = fmaxf(sum[j], 0.0f);
  *(v4f*)(out + (size_t)b * SIZE + s) = sum;
}

extern "C" void kernel_launch(void* const* d_in, const int* in_sizes, int n_in,
                              void* d_out, int out_size, void* d_ws, size_t ws_size,
                              hipStream_t stream) {
  const float* x    = (const float*)d_in[0];   // [32, 2048, 4096] f32
  const float* w    = (const float*)d_in[1];   // [2048, 4096] f32
  const float* bias = (const float*)d_in[2];   // [4096] f32
  float* out = (float*)d_out;                  // [32, 4096] f32

  const size_t need = (size_t)CSPLIT_MAX * BATCH * SIZE * sizeof(float); // 8 MiB
  const int csplit = (ws_size >= need) ? CSPLIT_MAX : 1;
  const int cpb = CH / csplit;

  dim3 grid(SIZE / S_PER_BLOCK, BATCH / BB, csplit);
  if (csplit > 1) {
    diag_dot_kernel<<<grid, TPB, 0, stream>>>(x, w, bias, (float*)d_ws, cpb, 0);
    const int n4 = BATCH * SIZE / 4;
    reduce_kernel<<<(n4 + 255) / 256, 256, 0, stream>>>(
        (const float*)d_ws, bias, out, csplit);
  } else {
    diag_dot_kernel<<<grid, TPB, 0, stream>>>(x, w, bias, out, cpb, 1);
  }
}